// HierarchicalGNN_72894184948036
// MI455X (gfx1250) — compile-verified
//
#include <hip/hip_runtime.h>
#include <limits>

#define N_MI 500000
#define E_MI 4000000
#define N_MA 35000
#define E_MA 560000
#define HID  64
#define LAT  32
#define HEADS 4
#define C1   (HEADS * HID)      // 256
#define E2T  (E_MA + N_MA)      // edges + self loops = 595000

typedef __attribute__((ext_vector_type(2))) float v2f;
typedef __attribute__((ext_vector_type(8))) float v8f;

// ---------------- helpers ----------------

__device__ __forceinline__ void atomicMaxF32(float* addr, float val) {
  // classic signed/unsigned trick; works with -inf init and NaN-free updates
  if (val >= 0.0f) atomicMax((int*)addr, __float_as_int(val));
  else             atomicMin((unsigned int*)addr, __float_as_uint(val));
}

__global__ void k_fill(float* __restrict__ p, float v, long long n) {
  long long i = (long long)blockIdx.x * blockDim.x + threadIdx.x;
  if (i < n) p[i] = v;
}

__global__ void k_count_dst(const int* __restrict__ dst, float* __restrict__ deg, int E) {
  int e = blockIdx.x * blockDim.x + threadIdx.x;
  if (e < E) atomicAdd(&deg[dst[e]], 1.0f);
}

__global__ void k_gcn_dinv(float* __restrict__ d, int n) {
  int i = blockIdx.x * blockDim.x + threadIdx.x;
  if (i < n) d[i] = rsqrtf(d[i] + 1.0f);   // deg + self loop
}

// ---------------- WMMA GEMM: C[M,N] = A[M,K] @ W[K,N] (+bias)(+relu) ----------------
// Requirements: N % 16 == 0 and (N/16) % TN == 0. All buffers < 2^31 elements
// (32-bit indexing). One wave computes a 16 x (TN*16) C tile via
// V_WMMA_F32_16X16X4_F32; A loaded once per K-step, feeds TN WMMAs.
// ISA layouts (CDNA5 ISA):
//   A 16x4 f32 : lanes 0-15 row M=p hold K=k0,k0+1 (v0,v1); lanes 16-31 K=k0+2,k0+3
//   B 4x16 f32 : row K striped across lanes (N=p), mirrored halves
//   C/D        : VGPR j, lanes 0-15 -> C[m0+j][n0+p], lanes 16-31 -> C[m0+8+j][n0+p]
// OOB rows are clamped for the load and zeroed with a select (no exec-mask
// branches inside the WMMA loop; EXEC stays all-1s as the ISA requires).
template<int TN>
__global__ void k_wmma_gemm(const float* __restrict__ A, const float* __restrict__ W,
                            const float* __restrict__ bias, float* __restrict__ C,
                            int M, int K, int N, int act) {
  int lane = threadIdx.x & 31;
  int wave = threadIdx.x >> 5;
  int tiles_n  = N >> 4;
  int groups_n = tiles_n / TN;
  int tiles_m  = (M + 15) >> 4;
  int tile = blockIdx.x * (blockDim.x >> 5) + wave;
  if (tile >= tiles_m * groups_n) return;   // uniform per wave
  int tm = tile / groups_n, tg = tile % groups_n;
  int m0 = tm << 4;
  int n0 = tg * (TN << 4);
  int half = lane >> 4, p = lane & 15;

  int  rowA = m0 + p;
  bool rok  = rowA < M;
  int  rowL = rok ? rowA : (M - 1);         // clamped address, value zeroed below
  const float* Arow = A + rowL * K;
  int  col0 = n0 + p;                       // always < N (N%16==0, groups exact)

  v8f acc[TN];
#pragma unroll
  for (int t = 0; t < TN; ++t) acc[t] = (v8f){0.f,0.f,0.f,0.f,0.f,0.f,0.f,0.f};

  int kmain = K & ~3;
  for (int k0 = 0; k0 < kmain; k0 += 4) {
    int ka = k0 + (half << 1);
    float a0 = Arow[ka];
    float a1 = Arow[ka + 1];
    a0 = rok ? a0 : 0.f;                    // v_cndmask, no branch
    a1 = rok ? a1 : 0.f;
    v2f a; a.x = a0; a.y = a1;
    const float* Wk = W + ka * N + col0;
#pragma unroll
    for (int t = 0; t < TN; ++t) {
      v2f b; b.x = Wk[t * 16]; b.y = Wk[N + t * 16];
      acc[t] = __builtin_amdgcn_wmma_f32_16x16x4_f32(false, a, false, b,
                                                     (short)0, acc[t], false, false);
    }
  }

  // K tail (K=17, K=34) as scalar FMAs in the accumulator layout, clamped loads
  for (int k = kmain; k < K; ++k) {
    float av[8];
#pragma unroll
    for (int j = 0; j < 8; ++j) {
      int r  = m0 + j + (half << 3);
      int rc = (r < M) ? r : (M - 1);
      float v = A[rc * K + k];
      av[j] = (r < M) ? v : 0.f;
    }
#pragma unroll
    for (int t = 0; t < TN; ++t) {
      float wv = W[k * N + col0 + t * 16];
#pragma unroll
      for (int j = 0; j < 8; ++j) acc[t][j] += av[j] * wv;
    }
  }

#pragma unroll
  for (int t = 0; t < TN; ++t) {
    int col = col0 + t * 16;
    float bv = (bias != nullptr) ? bias[col] : 0.f;
#pragma unroll
    for (int j = 0; j < 8; ++j) {
      int r = m0 + j + (half << 3);
      if (r < M) {
        float v = acc[t][j] + bv;
        if (act) v = fmaxf(v, 0.f);
        C[r * N + col] = v;
      }
    }
  }
}

// ---------------- GCN ----------------

__global__ void k_gcn_scatter(const float* __restrict__ h, const int* __restrict__ src,
                              const int* __restrict__ dst, const float* __restrict__ dinv,
                              float* __restrict__ out, long long E) {
  long long idx = (long long)blockIdx.x * blockDim.x + threadIdx.x;
  if (idx >= E * HID) return;
  int e = (int)(idx >> 6);
  int f = (int)(idx & 63);
  int s = src[e], d = dst[e];
  float w = dinv[s] * dinv[d];
  atomicAdd(&out[d * HID + f], h[s * HID + f] * w);
}

__global__ void k_gcn_finalize(const float* __restrict__ agg, float* __restrict__ hlin,
                               const float* __restrict__ dinv, const float* __restrict__ b,
                               int relu, long long n) {
  long long idx = (long long)blockIdx.x * blockDim.x + threadIdx.x;
  if (idx >= n) return;
  int i = (int)(idx >> 6);
  int f = (int)(idx & 63);
  float di = dinv[i];
  float v = agg[idx] + hlin[idx] * di * di + b[f];
  if (relu) v = fmaxf(v, 0.f);
  hlin[idx] = v;
}

// ---------------- pooling ----------------

__global__ void k_pool_scatter(const float* __restrict__ h, const int* __restrict__ batch,
                               float* __restrict__ mean, float* __restrict__ mx,
                               float* __restrict__ cnt, long long n) {
  long long idx = (long long)blockIdx.x * blockDim.x + threadIdx.x;
  if (idx >= n) return;
  int i = (int)(idx >> 6);
  int f = (int)(idx & 63);
  int b = batch[i];
  float v = h[idx];
  atomicAdd(&mean[b * HID + f], v);
  atomicMaxF32(&mx[b * HID + f], v);
  if (f == 0) atomicAdd(&cnt[b], 1.0f);
}

__global__ void k_build_z(const float* __restrict__ mean, const float* __restrict__ mx,
                          const float* __restrict__ cnt, float* __restrict__ z, int n) {
  int idx = blockIdx.x * blockDim.x + threadIdx.x;
  if (idx >= n * 2 * HID) return;
  int i = idx >> 7;
  int f = idx & 127;
  float v;
  if (f < HID) {
    v = mean[i * HID + f] / fmaxf(cnt[i], 1.0f);
  } else {
    float m = mx[i * HID + (f - HID)];
    v = (fabsf(m) < 3.0e38f) ? m : 0.0f;   // non-finite (-inf empty segment) -> 0
  }
  z[idx] = v;
}

__global__ void k_build_xc(const float* __restrict__ zm, const float* __restrict__ xmac,
                           float* __restrict__ xc, int n) {
  const int F = LAT + 2;  // 34
  int idx = blockIdx.x * blockDim.x + threadIdx.x;
  if (idx >= n * F) return;
  int i = idx / F;
  int f = idx % F;
  xc[idx] = (f < LAT) ? zm[i * LAT + f] : xmac[i * 2 + (f - LAT)];
}

// ---------------- GATv2 ----------------

__global__ void k_eattr_scatter(const float* __restrict__ eattr, const int* __restrict__ dst,
                                float* __restrict__ lattr, int E) {
  int idx = blockIdx.x * blockDim.x + threadIdx.x;
  if (idx >= E * 4) return;
  int e = idx >> 2, j = idx & 3;
  atomicAdd(&lattr[dst[e] * 4 + j], eattr[idx]);
}

__global__ void k_loop_div(float* __restrict__ lattr, const float* __restrict__ deg, int n) {
  int idx = blockIdx.x * blockDim.x + threadIdx.x;
  if (idx >= n * 4) return;
  lattr[idx] /= fmaxf(deg[idx >> 2], 1.0f);
}

// alpha[e,h] = sum_c leaky( xl[s,hc] + xr[d,hc] + (ea @ We)[hc] ) * att[h,c]
__global__ void k_gat_alpha(const float* __restrict__ xl, const float* __restrict__ xr,
                            const float* __restrict__ eattr, const float* __restrict__ lattr,
                            const int* __restrict__ src, const int* __restrict__ dst,
                            const float* __restrict__ We, const float* __restrict__ att,
                            float* __restrict__ alpha, int heads, int ch) {
  int idx = blockIdx.x * blockDim.x + threadIdx.x;
  if (idx >= E2T * heads) return;
  int e = idx / heads;
  int h = idx % heads;
  int s, d;
  const float* ea;
  if (e < E_MA) { s = src[e]; d = dst[e]; ea = eattr + e * 4; }
  else          { s = d = e - E_MA;       ea = lattr + s * 4; }
  int C = heads * ch;
  float ea0 = ea[0], ea1 = ea[1], ea2 = ea[2], ea3 = ea[3];
  const float* xls = xl + s * C + h * ch;
  const float* xrd = xr + d * C + h * ch;
  const float* w0  = We + h * ch;  // We row stride = C
  float sum = 0.f;
  for (int c = 0; c < ch; ++c) {
    float m = xls[c] + xrd[c] +
              ea0 * w0[c] + ea1 * w0[C + c] + ea2 * w0[2 * C + c] + ea3 * w0[3 * C + c];
    m = (m > 0.f) ? m : 0.2f * m;
    sum += m * att[h * ch + c];
  }
  alpha[idx] = sum;
}

__global__ void k_seg_max(const float* __restrict__ alpha, const int* __restrict__ dst,
                          float* __restrict__ amax, int heads) {
  int idx = blockIdx.x * blockDim.x + threadIdx.x;
  if (idx >= E2T * heads) return;
  int e = idx / heads, h = idx % heads;
  int d = (e < E_MA) ? dst[e] : (e - E_MA);
  atomicMaxF32(&amax[d * heads + h], alpha[idx]);
}

__global__ void k_exp_denom(float* __restrict__ alpha, const float* __restrict__ amax,
                            const int* __restrict__ dst, float* __restrict__ denom, int heads) {
  int idx = blockIdx.x * blockDim.x + threadIdx.x;
  if (idx >= E2T * heads) return;
  int e = idx / heads, h = idx % heads;
  int d = (e < E_MA) ? dst[e] : (e - E_MA);
  float ex = __expf(alpha[idx] - amax[d * heads + h]);
  alpha[idx] = ex;                 // overwrite alpha with exp(alpha - amax[d])
  atomicAdd(&denom[d * heads + h], ex);
}

__global__ void k_gat_agg(const float* __restrict__ xl, const float* __restrict__ ex,
                          const float* __restrict__ denom, const int* __restrict__ src,
                          const int* __restrict__ dst, float* __restrict__ out,
                          int heads, int ch) {
  int C = heads * ch;
  long long idx = (long long)blockIdx.x * blockDim.x + threadIdx.x;
  if (idx >= (long long)E2T * C) return;
  int e  = (int)(idx / C);
  int hc = (int)(idx % C);
  int h  = hc / ch;
  int s, d;
  if (e < E_MA) { s = src[e]; d = dst[e]; }
  else          { s = d = e - E_MA; }
  float a = ex[e * heads + h] / denom[d * heads + h];
  atomicAdd(&out[d * C + hc], xl[s * C + hc] * a);
}

__global__ void k_bias_elu(float* __restrict__ x, const float* __restrict__ b, long long n, int C) {
  long long idx = (long long)blockIdx.x * blockDim.x + threadIdx.x;
  if (idx >= n) return;
  int f = (int)(idx % C);
  float v = x[idx] + b[f];
  x[idx] = (v > 0.f) ? v : (__expf(v) - 1.0f);   // ELU, alpha=1
}

__global__ void k_final(const float* __restrict__ out2, const float* __restrict__ bias2,
                        const float* __restrict__ Wp, const float* __restrict__ bp,
                        float* __restrict__ y, int n) {
  int i = blockIdx.x * blockDim.x + threadIdx.x;
  if (i >= n) return;
  float s = 0.f;
#pragma unroll
  for (int c = 0; c < LAT; ++c) s += (out2[i * LAT + c] + bias2[c]) * Wp[c];
  y[i] = s + bp[0];
}

// ---------------- host orchestration ----------------

static inline unsigned gridFor(long long n, int b) { return (unsigned)((n + b - 1) / b); }

static void launch_gemm(const float* A, const float* W, const float* bias, float* C,
                        int M, int K, int N, int act, hipStream_t stream) {
  int tiles_m = (M + 15) / 16;
  int tiles_n = N / 16;                  // N % 16 == 0 for all call sites
  int TN = (tiles_n % 4 == 0) ? 4 : ((tiles_n % 2 == 0) ? 2 : 1);
  int tiles = tiles_m * (tiles_n / TN);
  unsigned blocks = (unsigned)((tiles + 7) / 8);   // 8 wave32 per block
  if (TN == 4)      k_wmma_gemm<4><<<blocks, 256, 0, stream>>>(A, W, bias, C, M, K, N, act);
  else if (TN == 2) k_wmma_gemm<2><<<blocks, 256, 0, stream>>>(A, W, bias, C, M, K, N, act);
  else              k_wmma_gemm<1><<<blocks, 256, 0, stream>>>(A, W, bias, C, M, K, N, act);
}

extern "C" void kernel_launch(void* const* d_in, const int* in_sizes, int n_in,
                              void* d_out, int out_size, void* d_ws, size_t ws_size,
                              hipStream_t stream) {
  (void)in_sizes; (void)n_in; (void)out_size; (void)ws_size;
  const float* x_micro = (const float*)d_in[0];
  const int*   ei_mi   = (const int*)d_in[1];
  const int*   src_mi  = ei_mi;
  const int*   dst_mi  = ei_mi + E_MI;
  const int*   batch   = (const int*)d_in[2];
  const float* x_macro = (const float*)d_in[3];
  const int*   ei_ma   = (const int*)d_in[4];
  const int*   src_ma  = ei_ma;
  const int*   dst_ma  = ei_ma + E_MA;
  const float* eattr   = (const float*)d_in[5];
  const float* W1  = (const float*)d_in[6];  const float* b1  = (const float*)d_in[7];
  const float* W2  = (const float*)d_in[8];  const float* b2  = (const float*)d_in[9];
  const float* Wm1 = (const float*)d_in[10]; const float* bm1 = (const float*)d_in[11];
  const float* Wm2 = (const float*)d_in[12]; const float* bm2 = (const float*)d_in[13];
  const float* Wl1 = (const float*)d_in[14]; const float* bl1 = (const float*)d_in[15];
  const float* Wr1 = (const float*)d_in[16]; const float* br1 = (const float*)d_in[17];
  const float* We1 = (const float*)d_in[18]; const float* att1 = (const float*)d_in[19];
  const float* bias1 = (const float*)d_in[20];
  const float* Wl2 = (const float*)d_in[21]; const float* bl2 = (const float*)d_in[22];
  const float* Wr2 = (const float*)d_in[23]; const float* br2 = (const float*)d_in[24];
  const float* We2 = (const float*)d_in[25]; const float* att2 = (const float*)d_in[26];
  const float* bias2 = (const float*)d_in[27];
  const float* Wp  = (const float*)d_in[28]; const float* bp  = (const float*)d_in[29];

  char* ws = (char*)d_ws;
  const size_t BIG = (size_t)N_MI * HID * sizeof(float);     // 128 MB
  float* bufA = (float*)ws;
  float* bufB = (float*)(ws + BIG);
  size_t off = 2 * BIG;
  auto take = [&](size_t elems) {
    float* p = (float*)(ws + off);
    off += ((elems * sizeof(float)) + 255) & ~(size_t)255;
    return p;
  };
  float* dinv_mi = take(N_MI);
  float* pmean   = take((size_t)N_MA * HID);
  float* pmax    = take((size_t)N_MA * HID);
  float* pcnt    = take(N_MA);
  float* zbuf    = take((size_t)N_MA * 2 * HID);

  // Macro-phase buffers: reuse the 256MB micro region (dead after pooling scatter)
  size_t moff = 0;
  auto mtake = [&](size_t elems) {
    float* p = (float*)(ws + moff);
    moff += ((elems * sizeof(float)) + 255) & ~(size_t)255;
    return p;
  };
  float* t1      = mtake((size_t)N_MA * HID);
  float* zmorpho = mtake((size_t)N_MA * LAT);
  float* xc      = mtake((size_t)N_MA * (LAT + 2));
  float* xl1     = mtake((size_t)N_MA * C1);
  float* xr1     = mtake((size_t)N_MA * C1);
  float* deg_ma  = mtake(N_MA);
  float* lattr   = mtake((size_t)N_MA * 4);
  float* alpha1  = mtake((size_t)E2T * HEADS);
  float* amax1   = mtake((size_t)N_MA * HEADS);
  float* den1    = mtake((size_t)N_MA * HEADS);
  float* out1    = mtake((size_t)N_MA * C1);    // -> h2 after bias+ELU
  float* xl2     = mtake((size_t)N_MA * LAT);
  float* xr2     = mtake((size_t)N_MA * LAT);
  float* alpha2  = mtake(E2T);
  float* amax2   = mtake(N_MA);
  float* den2    = mtake(N_MA);
  float* out2    = mtake((size_t)N_MA * LAT);

  const int B = 256;
  const float NEG_INF = -std::numeric_limits<float>::infinity();
  const long long NH = (long long)N_MI * HID;

  // --- micro degree -> dinv (shared by both GCN layers) ---
  k_fill<<<gridFor(N_MI, B), B, 0, stream>>>(dinv_mi, 0.f, N_MI);
  k_count_dst<<<gridFor(E_MI, B), B, 0, stream>>>(dst_mi, dinv_mi, E_MI);
  k_gcn_dinv<<<gridFor(N_MI, B), B, 0, stream>>>(dinv_mi, N_MI);

  // --- GCN layer 1: bufA = x@W1 ; bufB = scatter ; bufA = relu(agg + self + b1) ---
  launch_gemm(x_micro, W1, nullptr, bufA, N_MI, 17, HID, 0, stream);
  k_fill<<<gridFor(NH, B), B, 0, stream>>>(bufB, 0.f, NH);
  k_gcn_scatter<<<gridFor((long long)E_MI * HID, B), B, 0, stream>>>(bufA, src_mi, dst_mi, dinv_mi, bufB, E_MI);
  k_gcn_finalize<<<gridFor(NH, B), B, 0, stream>>>(bufB, bufA, dinv_mi, b1, 1, NH);

  // --- GCN layer 2: bufB = bufA@W2 ; bufA = scatter ; bufB = agg + self + b2 ---
  launch_gemm(bufA, W2, nullptr, bufB, N_MI, HID, HID, 0, stream);
  k_fill<<<gridFor(NH, B), B, 0, stream>>>(bufA, 0.f, NH);
  k_gcn_scatter<<<gridFor((long long)E_MI * HID, B), B, 0, stream>>>(bufB, src_mi, dst_mi, dinv_mi, bufA, E_MI);
  k_gcn_finalize<<<gridFor(NH, B), B, 0, stream>>>(bufA, bufB, dinv_mi, b2, 0, NH);  // final h in bufB

  // --- pooling (mean + max + count) ---
  k_fill<<<gridFor((long long)N_MA * HID, B), B, 0, stream>>>(pmean, 0.f, (long long)N_MA * HID);
  k_fill<<<gridFor((long long)N_MA * HID, B), B, 0, stream>>>(pmax, NEG_INF, (long long)N_MA * HID);
  k_fill<<<gridFor(N_MA, B), B, 0, stream>>>(pcnt, 0.f, N_MA);
  k_pool_scatter<<<gridFor(NH, B), B, 0, stream>>>(bufB, batch, pmean, pmax, pcnt, NH);
  k_build_z<<<gridFor((long long)N_MA * 2 * HID, B), B, 0, stream>>>(pmean, pmax, pcnt, zbuf, N_MA);

  // --- morpho MLP ---
  launch_gemm(zbuf, Wm1, bm1, t1, N_MA, 2 * HID, HID, 1, stream);
  launch_gemm(t1, Wm2, bm2, zmorpho, N_MA, HID, LAT, 0, stream);
  k_build_xc<<<gridFor((long long)N_MA * (LAT + 2), B), B, 0, stream>>>(zmorpho, x_macro, xc, N_MA);

  // --- macro degree + mean self-loop edge attr (shared by both GAT layers) ---
  k_fill<<<gridFor(N_MA, B), B, 0, stream>>>(deg_ma, 0.f, N_MA);
  k_count_dst<<<gridFor(E_MA, B), B, 0, stream>>>(dst_ma, deg_ma, E_MA);
  k_fill<<<gridFor((long long)N_MA * 4, B), B, 0, stream>>>(lattr, 0.f, (long long)N_MA * 4);
  k_eattr_scatter<<<gridFor((long long)E_MA * 4, B), B, 0, stream>>>(eattr, dst_ma, lattr, E_MA);
  k_loop_div<<<gridFor((long long)N_MA * 4, B), B, 0, stream>>>(lattr, deg_ma, N_MA);

  // --- GATv2 layer 1 (heads=4, ch=64) ---
  launch_gemm(xc, Wl1, bl1, xl1, N_MA, LAT + 2, C1, 0, stream);
  launch_gemm(xc, Wr1, br1, xr1, N_MA, LAT + 2, C1, 0, stream);
  k_gat_alpha<<<gridFor((long long)E2T * HEADS, B), B, 0, stream>>>(xl1, xr1, eattr, lattr, src_ma, dst_ma, We1, att1, alpha1, HEADS, HID);
  k_fill<<<gridFor((long long)N_MA * HEADS, B), B, 0, stream>>>(amax1, NEG_INF, (long long)N_MA * HEADS);
  k_seg_max<<<gridFor((long long)E2T * HEADS, B), B, 0, stream>>>(alpha1, dst_ma, amax1, HEADS);
  k_fill<<<gridFor((long long)N_MA * HEADS, B), B, 0, stream>>>(den1, 0.f, (long long)N_MA * HEADS);
  k_exp_denom<<<gridFor((long long)E2T * HEADS, B), B, 0, stream>>>(alpha1, amax1, dst_ma, den1, HEADS);
  k_fill<<<gridFor((long long)N_MA * C1, B), B, 0, stream>>>(out1, 0.f, (long long)N_MA * C1);
  k_gat_agg<<<gridFor((long long)E2T * C1, B), B, 0, stream>>>(xl1, alpha1, den1, src_ma, dst_ma, out1, HEADS, HID);
  k_bias_elu<<<gridFor((long long)N_MA * C1, B), B, 0, stream>>>(out1, bias1, (long long)N_MA * C1, C1);

  // --- GATv2 layer 2 (heads=1, ch=32) ---
  launch_gemm(out1, Wl2, bl2, xl2, N_MA, C1, LAT, 0, stream);
  launch_gemm(out1, Wr2, br2, xr2, N_MA, C1, LAT, 0, stream);
  k_gat_alpha<<<gridFor(E2T, B), B, 0, stream>>>(xl2, xr2, eattr, lattr, src_ma, dst_ma, We2, att2, alpha2, 1, LAT);
  k_fill<<<gridFor(N_MA, B), B, 0, stream>>>(amax2, NEG_INF, N_MA);
  k_seg_max<<<gridFor(E2T, B), B, 0, stream>>>(alpha2, dst_ma, amax2, 1);
  k_fill<<<gridFor(N_MA, B), B, 0, stream>>>(den2, 0.f, N_MA);
  k_exp_denom<<<gridFor(E2T, B), B, 0, stream>>>(alpha2, amax2, dst_ma, den2, 1);
  k_fill<<<gridFor((long long)N_MA * LAT, B), B, 0, stream>>>(out2, 0.f, (long long)N_MA * LAT);
  k_gat_agg<<<gridFor((long long)E2T * LAT, B), B, 0, stream>>>(xl2, alpha2, den2, src_ma, dst_ma, out2, 1, LAT);

  // --- final head: (out2 + bias2) @ Wp + bp ---
  k_final<<<gridFor(N_MA, B), B, 0, stream>>>(out2, bias2, Wp, bp, (float*)d_out, N_MA);
}